// SuperLinear_54168127537255
// MI455X (gfx1250) — compile-verified
//
#include <hip/hip_runtime.h>

// out[b,n] = sum_m x[b,n,m] * w1[m,0,n] + b1[n]
// B=256, N=4096, M=64.  Memory-bound (AI ~= 0.5 flop/byte); stream x once.
// WMMA mapping: per 16x16 (b,n) tile, for each j: D_j = X(16x64 @ n0+j) * W(64x16)
// via 16 chained V_WMMA_F32_16X16X4_F32; keep column j (v_cndmask select).

typedef __attribute__((ext_vector_type(2))) float v2f;
typedef __attribute__((ext_vector_type(8))) float v8f;

#define N_DIM 4096
#define M_DIM 64

__global__ __launch_bounds__(256) void superlinear_wmma_f32(
    const float* __restrict__ x,    // [256, 4096, 64]
    const float* __restrict__ w1,   // [64, 1, 4096]
    const float* __restrict__ b1,   // [1, 4096, 1]
    float* __restrict__ out)        // [256, 4096]
{
    const int lane  = threadIdx.x & 31;
    const int waveI = threadIdx.x >> 5;
    const int tile  = blockIdx.x * 8 + waveI;   // 0..4095 tiles
    const int bTile = tile & 15;                // 16 tiles over B=256
    const int nGrp  = tile >> 4;                // 256 groups over N=4096
    const int b0    = bTile << 4;
    const int n0    = nGrp  << 4;
    const int L     = lane & 15;                // column / M-row lane id
    const int h     = lane >> 4;                // lane half selects K sub-pair

    // --- B fragments: w tile [64 x 16], reused for all 16 j. 32 VGPRs. ---
    // B 4x16 f32 layout: VGPR0 = row K=2h, VGPR1 = row K=2h+1 (N = lane&15).
    v2f bw[16];
#pragma unroll
    for (int k = 0; k < 16; ++k) {
        const int m = 4 * k + 2 * h;
        bw[k].x = w1[(size_t)m       * N_DIM + n0 + L];
        bw[k].y = w1[(size_t)(m + 1) * N_DIM + n0 + L];
    }

    const float bias = b1[n0 + L];

    // Per-lane x base: row b0+L, n-group start, this lane's K sub-pair.
    const float* px_base =
        x + ((size_t)(b0 + L) * N_DIM + n0) * M_DIM + 2 * h;

    v8f res = {};

#pragma unroll 1
    for (int j = 0; j < 16; ++j) {
        const float* px = px_base + (size_t)j * M_DIM;
        if (j < 15) __builtin_prefetch(px + M_DIM, 0, 0);  // global_prefetch_b8

        // A fragments for this n: 16 K-steps, lane holds (K=2h, 2h+1) as float2.
        v2f a[16];
#pragma unroll
        for (int k = 0; k < 16; ++k)
            a[k] = *(const v2f*)(px + 4 * k);

        // Two accumulator chains; bias folded into cA (C lanes carry N=L).
        v8f cA, cB = {};
#pragma unroll
        for (int r = 0; r < 8; ++r) cA[r] = bias;

#pragma unroll
        for (int k = 0; k < 16; k += 2) {
            cA = __builtin_amdgcn_wmma_f32_16x16x4_f32(
                false, a[k],     false, bw[k],     (short)0, cA, false, false);
            cB = __builtin_amdgcn_wmma_f32_16x16x4_f32(
                false, a[k + 1], false, bw[k + 1], (short)0, cB, false, false);
        }

        const v8f c = cA + cB;

        // Only column j of D_j is the true result: out[b0+r+8h, n0+j].
        const bool sel = (L == j);
#pragma unroll
        for (int r = 0; r < 8; ++r)
            res[r] = sel ? c[r] : res[r];
    }

    // res now holds the full 16x16 tile in standard C/D layout:
    // VGPR r, lane (h,L) -> out[b0 + r + 8h, n0 + L].  Coalesced stores.
#pragma unroll
    for (int r = 0; r < 8; ++r)
        out[(size_t)(b0 + r + 8 * h) * N_DIM + n0 + L] = res[r];
}

extern "C" void kernel_launch(void* const* d_in, const int* in_sizes, int n_in,
                              void* d_out, int out_size, void* d_ws, size_t ws_size,
                              hipStream_t stream) {
    (void)in_sizes; (void)n_in; (void)d_ws; (void)ws_size; (void)out_size;
    const float* x  = (const float*)d_in[0];
    const float* w1 = (const float*)d_in[1];
    const float* b1 = (const float*)d_in[2];
    float* out      = (float*)d_out;

    // 4096 tiles of 16x16; 8 waves (256 threads) per block -> 512 blocks.
    superlinear_wmma_f32<<<512, 256, 0, stream>>>(x, w1, b1, out);
}